// GNN_GBNeck_13563506720830
// MI455X (gfx1250) — compile-verified
//
#include <hip/hip_runtime.h>
#include <hip/hip_bf16.h>

// ---------------------------------------------------------------------------
// GB-Neck energies + forces for MI455X (gfx1250, wave32).
// Pipeline:
//   k_zero       : zero energy/force outputs and the gB scratch
//   k_neighbors  : per-molecule d^2 via V_WMMA_F32_16X16X4_F32 tiles in LDS
//                  (4 independent WMMA tiles per wave, fully unrolled),
//                  then wave32 shuffle top-K(32) under cutoff -> edge list
//   k_born       : HCT integral -> Born radii B, plus dB/dpsi*rho for bwd
//   k_energy     : GB pair + self energies, dE/dd and dE/dB grads, forces
//   k_force_born : chain dS/dB -> dI/dd -> forces through the Born integral
// Gather loops prefetch the next neighbor's position (global_prefetch_b8)
// to hide L2 latency in these latency-bound kernels.
// ---------------------------------------------------------------------------

#define NMOL   64
#define NA     512
#define NTOT   (NMOL * NA)
#define KNB    32
#define OFFS   0.009f
#define R2CUT  0.16f                 // RADIUS^2
#define ALPHA_ 1.0f
#define BETA_  0.8f
#define GAMMA_ 4.85f
#define CPREF  (-0.5f * 138.935456f * (1.0f - 1.0f / 78.5f))  // -0.5*KE*EPS_FACT

typedef float v2f __attribute__((ext_vector_type(2)));
typedef float v8f __attribute__((ext_vector_type(8)));

#define SLAB_STRIDE 520   // 512 + pad to break LDS bank alignment

// ---------------------------------------------------------------------------
// Kernel 0: zero outputs + gB scratch
// ---------------------------------------------------------------------------
__global__ __launch_bounds__(256) void k_zero(float* out, float* gB) {
    int i = blockIdx.x * 256 + threadIdx.x;
    if (i < NMOL + NTOT * 3) out[i] = 0.0f;
    if (i < NTOT)            gB[i]  = 0.0f;
}

// ---------------------------------------------------------------------------
// Kernel 1: neighbor build.
// grid = NMOL * 32 blocks (one per 16-row tile), 256 threads (8 waves).
// Phase 1: each wave computes 4 of the 32 column tiles of the 16x512 d^2
//          slab with V_WMMA_F32_16X16X4_F32:
//            A row i = (-2x_i, -2y_i, -2z_i, 1)      (16x4, K padded)
//            B col j = ( x_j ,  y_j ,  z_j , |p_j|^2) (4x16)
//            C row i = |p_i|^2 broadcast             (16x16)
//          => D = d^2 tile directly, one WMMA per tile, 4 independent
//          tiles unrolled for XDL/LDS latency hiding.
// Phase 2: per-row top-K(32) under cutoff via packed u64 shfl_xor argmin.
// ---------------------------------------------------------------------------
__global__ __launch_bounds__(256) void k_neighbors(const float* __restrict__ pos,
                                                   int* __restrict__ nbr) {
    __shared__ float lp[NA * 4];                 // x,y,z,|p|^2 per atom
    __shared__ float slab[16 * SLAB_STRIDE];     // 16 rows x 512 cols of d^2

    const int mol  = blockIdx.x >> 5;
    const int rt   = blockIdx.x & 31;
    const int tid  = threadIdx.x;
    const int base = mol * NA;

    for (int a = tid; a < NA; a += 256) {
        float x = pos[(base + a) * 3 + 0];
        float y = pos[(base + a) * 3 + 1];
        float z = pos[(base + a) * 3 + 2];
        lp[a * 4 + 0] = x; lp[a * 4 + 1] = y; lp[a * 4 + 2] = z;
        lp[a * 4 + 3] = x * x + y * y + z * z;
    }
    __syncthreads();

    const int wave = tid >> 5;
    const int lane = tid & 31;
    const int r0   = rt * 16;
    const int half = (lane < 16) ? 0 : 1;       // which K-pair / M-offset half
    const int l16  = lane & 15;
    const int mo   = half ? 8 : 0;

    // A matrix (32-bit 16x4 layout, ISA 7.12.2):
    //   lanes 0-15 : VGPR0=K0, VGPR1=K1 for M=lane
    //   lanes 16-31: VGPR0=K2, VGPR1=K3 for M=lane-16
    v2f amat;
    if (!half) { amat.x = -2.0f * lp[(r0 + l16) * 4 + 0];
                 amat.y = -2.0f * lp[(r0 + l16) * 4 + 1]; }
    else       { amat.x = -2.0f * lp[(r0 + l16) * 4 + 2];
                 amat.y = 1.0f; }

    // C matrix: VGPR v holds M=v (lanes 0-15) / M=v+8 (lanes 16-31), any N.
    v8f cinit;
#pragma unroll
    for (int v = 0; v < 8; ++v) cinit[v] = lp[(r0 + v + mo) * 4 + 3];

    // 4 independent column tiles per wave, fully unrolled: all B-loads can
    // be hoisted, 4 WMMAs issue back-to-back with no loop-carried deps.
#pragma unroll
    for (int t = 0; t < 4; ++t) {
        const int ct = wave + t * 8;
        const int c0 = ct * 16;
        v8f dtile;
#if __has_builtin(__builtin_amdgcn_wmma_f32_16x16x4_f32)
        // B matrix (4x16): rows striped across lanes within a VGPR,
        // K split across lane halves like C/D striping.
        v2f bmat;
        if (!half) { bmat.x = lp[(c0 + l16) * 4 + 0];       // K0 = x_j
                     bmat.y = lp[(c0 + l16) * 4 + 1]; }     // K1 = y_j
        else       { bmat.x = lp[(c0 + l16) * 4 + 2];       // K2 = z_j
                     bmat.y = lp[(c0 + l16) * 4 + 3]; }     // K3 = |p_j|^2
        dtile = __builtin_amdgcn_wmma_f32_16x16x4_f32(
            false, amat, false, bmat, (short)0, cinit, false, false);
#else
#pragma unroll
        for (int v = 0; v < 8; ++v) {
            int ri = r0 + v + mo, cj = c0 + l16;
            float dx = lp[ri * 4 + 0] - lp[cj * 4 + 0];
            float dy = lp[ri * 4 + 1] - lp[cj * 4 + 1];
            float dz = lp[ri * 4 + 2] - lp[cj * 4 + 2];
            dtile[v] = dx * dx + dy * dy + dz * dz;
        }
#endif
        // Scatter to slab; exclude self-pairs (reference adds 1e6*I to d^2).
#pragma unroll
        for (int v = 0; v < 8; ++v) {
            int row = v + mo;
            int col = c0 + l16;
            float d2 = dtile[v];
            if (r0 + row == col) d2 = 1.0e6f;
            slab[row * SLAB_STRIDE + col] = d2;
        }
    }
    __syncthreads();

    // Phase 2: top-K per row. Wave w owns rows 2w, 2w+1.
    for (int rr = 0; rr < 2; ++rr) {
        const int row  = wave * 2 + rr;
        const int atom = base + r0 + row;
        float vals[16];
#pragma unroll
        for (int t = 0; t < 16; ++t)
            vals[t] = slab[row * SLAB_STRIDE + (t * 32 + lane)];

        for (int k = 0; k < KNB; ++k) {
            float bv = vals[0]; int bt = 0;
#pragma unroll
            for (int t = 1; t < 16; ++t)
                if (vals[t] < bv) { bv = vals[t]; bt = t; }
            // pack (d2 bits || column); non-negative f32 orders as u32
            unsigned long long key =
                (((unsigned long long)__float_as_uint(bv)) << 32) |
                (unsigned)(bt * 32 + lane);
#pragma unroll
            for (int m = 16; m >= 1; m >>= 1) {
                unsigned long long o = __shfl_xor(key, m, 32);
                if (o < key) key = o;
            }
            float mv   = __uint_as_float((unsigned)(key >> 32));
            int   mcol = (int)(key & 0xffffffffu);
            if (lane == 0)
                nbr[atom * KNB + k] = (mv < R2CUT) ? (base + mcol) : -1;
            // invalidate winner's register copy (select chain, no scratch)
            bool own = ((mcol & 31) == lane);
            int  ot  = mcol >> 5;
#pragma unroll
            for (int t = 0; t < 16; ++t)
                vals[t] = (own && t == ot) ? 3.0e38f : vals[t];
        }
    }
}

// ---------------------------------------------------------------------------
// Kernel 2: HCT integral -> Born radii. Stores B[i] and G[i]=dB/dpsi*rho_i.
// ---------------------------------------------------------------------------
__global__ __launch_bounds__(256) void k_born(const float* __restrict__ pos,
                                              const float* __restrict__ gb,
                                              const int* __restrict__ nbr,
                                              float* __restrict__ Bout,
                                              float* __restrict__ Gout) {
    int i = blockIdx.x * 256 + threadIdx.x;
    if (i >= NTOT) return;
    int   li    = i & (NA - 1);
    float radi  = gb[li * 3 + 1];
    float rho_i = radi - OFFS;
    float xi = pos[i * 3 + 0], yi = pos[i * 3 + 1], zi = pos[i * 3 + 2];

    float Isum = 0.0f;
    for (int k = 0; k < KNB; ++k) {
        int j = nbr[i * KNB + k];
        if (j < 0) break;                         // sorted: invalids are suffix
        if (k + 1 < KNB) {                        // hide next gather's latency
            int jn = nbr[i * KNB + k + 1];
            if (jn >= 0) __builtin_prefetch(&pos[jn * 3], 0, 3);
        }
        int   lj = j & (NA - 1);
        float dx = pos[j * 3 + 0] - xi;
        float dy = pos[j * 3 + 1] - yi;
        float dz = pos[j * 3 + 2] - zi;
        float d   = sqrtf(dx * dx + dy * dy + dz * dz + 1e-12f);
        float srj = gb[lj * 3 + 2] * (gb[lj * 3 + 1] - OFFS);
        float U    = d + srj;
        float L    = fmaxf(fabsf(d - srj), rho_i);
        float invU = 1.0f / U, invL = 1.0f / L;
        float I = 0.5f * (invL - invU +
                          0.25f * (d - srj * srj / d) * (invU * invU - invL * invL) +
                          0.5f * logf(L * invU) / d);
        I = (rho_i < U) ? I : 0.0f;
        if (rho_i < srj - d) I += 2.0f * (1.0f / rho_i - invL);
        Isum += I;
    }
    float psi = Isum * rho_i;
    float u   = psi * (ALPHA_ + psi * (-BETA_ + psi * GAMMA_));
    float th  = tanhf(u);
    float Bi  = 1.0f / (1.0f / rho_i - th / radi);
    float dud = ALPHA_ - 2.0f * BETA_ * psi + 3.0f * GAMMA_ * psi * psi;
    Bout[i] = Bi;
    Gout[i] = Bi * Bi * (1.0f - th * th) * dud / radi * rho_i;  // dB/dIsum
}

// ---------------------------------------------------------------------------
// Kernel 3: pair + self energies, dE/dB scatter, direct dE/dd forces,
//           per-molecule energy accumulation.
// ---------------------------------------------------------------------------
__global__ __launch_bounds__(256) void k_energy(const float* __restrict__ pos,
                                                const float* __restrict__ gb,
                                                const int* __restrict__ nbr,
                                                const float* __restrict__ B,
                                                float* __restrict__ gB,
                                                float* __restrict__ out_energy,
                                                float* __restrict__ out_force) {
    int i = blockIdx.x * 256 + threadIdx.x;
    if (i >= NTOT) return;
    int   li = i & (NA - 1);
    float qi = gb[li * 3 + 0];
    float Bi = B[i];
    float xi = pos[i * 3 + 0], yi = pos[i * 3 + 1], zi = pos[i * 3 + 2];

    float epair = 0.0f;
    float gBi   = -CPREF * qi * qi / (Bi * Bi);  // d/dB of self term c*q^2/B
    float fx = 0.0f, fy = 0.0f, fz = 0.0f;

    for (int k = 0; k < KNB; ++k) {
        int j = nbr[i * KNB + k];
        if (j < 0) break;
        if (k + 1 < KNB) {
            int jn = nbr[i * KNB + k + 1];
            if (jn >= 0) { __builtin_prefetch(&pos[jn * 3], 0, 3);
                           __builtin_prefetch(&B[jn], 0, 3); }
        }
        int   lj = j & (NA - 1);
        float qj = gb[lj * 3 + 0];
        float Bj = B[j];
        float dx = pos[j * 3 + 0] - xi;
        float dy = pos[j * 3 + 1] - yi;
        float dz = pos[j * 3 + 2] - zi;
        float d2  = dx * dx + dy * dy + dz * dz + 1e-12f;
        float d   = sqrtf(d2);
        float Bij = Bi * Bj;
        float t   = expf(-d2 / (4.0f * Bij));
        float f2  = d2 + Bij * t;
        float f   = sqrtf(f2);
        epair += qi * qj / f;
        float common = -CPREF * qi * qj / (2.0f * f2 * f);   // -c*qq/(2 f^3)
        float dEdd = common * (2.0f * d * (1.0f - 0.25f * t));
        float dEdB = common * t * (1.0f + d2 / (4.0f * Bij)); // w.r.t. Bij
        gBi += dEdB * Bj;
        atomicAdd(&gB[j], dEdB * Bi);
        float s = dEdd / d;            // F_i = +dE/dd * dvec/d (force=-grad)
        fx += s * dx; fy += s * dy; fz += s * dz;
        atomicAdd(&out_force[j * 3 + 0], -s * dx);
        atomicAdd(&out_force[j * 3 + 1], -s * dy);
        atomicAdd(&out_force[j * 3 + 2], -s * dz);
    }
    atomicAdd(&gB[i], gBi);
    atomicAdd(&out_force[i * 3 + 0], fx);
    atomicAdd(&out_force[i * 3 + 1], fy);
    atomicAdd(&out_force[i * 3 + 2], fz);
    atomicAdd(&out_energy[i >> 9], CPREF * (epair + qi * qi / Bi));
}

// ---------------------------------------------------------------------------
// Kernel 4: backward through the HCT integral: dS/dd = gB[i]*G[i]*dI/dd.
// ---------------------------------------------------------------------------
__global__ __launch_bounds__(256) void k_force_born(const float* __restrict__ pos,
                                                    const float* __restrict__ gb,
                                                    const int* __restrict__ nbr,
                                                    const float* __restrict__ gB,
                                                    const float* __restrict__ G,
                                                    float* __restrict__ out_force) {
    int i = blockIdx.x * 256 + threadIdx.x;
    if (i >= NTOT) return;
    int   li    = i & (NA - 1);
    float rho_i = gb[li * 3 + 1] - OFFS;
    float gI    = gB[i] * G[i];
    float xi = pos[i * 3 + 0], yi = pos[i * 3 + 1], zi = pos[i * 3 + 2];
    float fx = 0.0f, fy = 0.0f, fz = 0.0f;

    for (int k = 0; k < KNB; ++k) {
        int j = nbr[i * KNB + k];
        if (j < 0) break;
        if (k + 1 < KNB) {
            int jn = nbr[i * KNB + k + 1];
            if (jn >= 0) __builtin_prefetch(&pos[jn * 3], 0, 3);
        }
        int   lj = j & (NA - 1);
        float dx = pos[j * 3 + 0] - xi;
        float dy = pos[j * 3 + 1] - yi;
        float dz = pos[j * 3 + 2] - zi;
        float d  = sqrtf(dx * dx + dy * dy + dz * dz + 1e-12f);
        float s  = gb[lj * 3 + 2] * (gb[lj * 3 + 1] - OFFS);
        float U = d + s, invU = 1.0f / U;
        float am = d - s, aam = fabsf(am);
        bool  lclamp = aam > rho_i;
        float L    = lclamp ? aam : rho_i;
        float invL = 1.0f / L;
        float dL   = lclamp ? ((am >= 0.0f) ? 1.0f : -1.0f) : 0.0f;
        float invU2 = invU * invU, invL2 = invL * invL;
        float s2 = s * s;
        float dIdd = 0.0f;
        if (rho_i < U) {
            float term = -dL * invL2 + invU2;
            term += 0.25f * ((1.0f + s2 / (d * d)) * (invU2 - invL2) +
                             (d - s2 / d) * (-2.0f * invU2 * invU +
                                              2.0f * dL * invL2 * invL));
            term += 0.5f * ((dL * invL - invU) / d - logf(L * invU) / (d * d));
            dIdd = 0.5f * term;
        }
        if (rho_i < s - d) dIdd += 2.0f * dL * invL2;
        float sc = gI * dIdd / d;
        fx += sc * dx; fy += sc * dy; fz += sc * dz;
        atomicAdd(&out_force[j * 3 + 0], -sc * dx);
        atomicAdd(&out_force[j * 3 + 1], -sc * dy);
        atomicAdd(&out_force[j * 3 + 2], -sc * dz);
    }
    atomicAdd(&out_force[i * 3 + 0], fx);
    atomicAdd(&out_force[i * 3 + 1], fy);
    atomicAdd(&out_force[i * 3 + 2], fz);
}

// ---------------------------------------------------------------------------
extern "C" void kernel_launch(void* const* d_in, const int* in_sizes, int n_in,
                              void* d_out, int out_size, void* d_ws, size_t ws_size,
                              hipStream_t stream) {
    const float* pos = (const float*)d_in[0];       // [N,3]
    // d_in[1] = batch (implicit: i>>9), d_in[2] = gbparameters [512,3]
    const float* gb  = (const float*)d_in[2];

    float* out        = (float*)d_out;              // [64] energy ++ [N,3] forces
    float* out_energy = out;
    float* out_force  = out + NMOL;

    int*   nbr = (int*)d_ws;                        // N*K ints
    float* B   = (float*)(nbr + (size_t)NTOT * KNB);
    float* Gd  = B + NTOT;
    float* gB  = Gd + NTOT;

    k_zero<<<(NMOL + NTOT * 3 + 255) / 256, 256, 0, stream>>>(out, gB);
    k_neighbors<<<NMOL * 32, 256, 0, stream>>>(pos, nbr);
    k_born<<<NTOT / 256, 256, 0, stream>>>(pos, gb, nbr, B, Gd);
    k_energy<<<NTOT / 256, 256, 0, stream>>>(pos, gb, nbr, B, gB, out_energy, out_force);
    k_force_born<<<NTOT / 256, 256, 0, stream>>>(pos, gb, nbr, gB, Gd, out_force);
}